// GHMC_61873298866306
// MI455X (gfx1250) — compile-verified
//
#include <hip/hip_runtime.h>
#include <hip/hip_bf16.h>
#include <math.h>

// GHM-C loss, single pass + WMMA-based finalize reduction.
// loss = (1/n) * sum_b S_b / counts_b   (tot cancels algebraically)

typedef __attribute__((ext_vector_type(2))) float v2f;
typedef __attribute__((ext_vector_type(8))) float v8f;

#define GHMC_BINS 10
#define NSLOT 20            // 2*BINS: even slot = count, odd slot = bce-sum
#define WAVES_PER_BLOCK 8
#define TPB 256

// ---------------------------------------------------------------------------
// Per-element math. `base` = &hist[wave][0][lane]; slot s lives at base+s*32,
// so all 32 lanes of a wave always touch 32 distinct LDS banks (no conflicts,
// no cross-lane contention -> ds_add_f32 runs at full rate, deterministically).
// ---------------------------------------------------------------------------
__device__ __forceinline__ void ghmc_elem(float x, float t, float w,
                                          float* __restrict__ base) {
  const bool valid = w > 0.0f;
  const float s = 1.0f / (1.0f + __expf(-x));            // sigmoid
  const float g = fabsf(s - t);
  int b = (int)(g * 10.0f);                              // trunc, matches jnp
  b = (b > GHMC_BINS - 1) ? (GHMC_BINS - 1) : (b < 0 ? 0 : b);
  // stable logaddexp(0, x) - x*t
  const float bce = fmaxf(x, 0.0f) + __logf(1.0f + __expf(-fabsf(x))) - x * t;
  if (valid) {
    atomicAdd(base + (2 * b) * 32, 1.0f);                // ds_add_f32
    atomicAdd(base + (2 * b + 1) * 32, bce);             // ds_add_f32
  }
}

// ---------------------------------------------------------------------------
// Pass 1: stream pred/target/label_weight with b128 loads, per-lane LDS
// histogram, emit 32 floats of block partials (slots 20..31 zero-padded so
// the WMMA finalize can consume a clean [nb x 32] f32 matrix).
// ---------------------------------------------------------------------------
__global__ void __launch_bounds__(TPB) ghmc_hist_kernel(
    const float4* __restrict__ pred4, const float4* __restrict__ tgt4,
    const float4* __restrict__ lw4, float* __restrict__ partials,
    const float* __restrict__ pred_s, const float* __restrict__ tgt_s,
    const float* __restrict__ lw_s, int n4, int n_tail_begin, int n) {
  __shared__ float hist[WAVES_PER_BLOCK][NSLOT][32];
  const int tid = threadIdx.x;
  const int lane = tid & 31;
  const int wave = tid >> 5;
  float* base = &hist[wave][0][lane];

  for (int i = tid; i < WAVES_PER_BLOCK * NSLOT * 32; i += TPB)
    ((float*)hist)[i] = 0.0f;
  __syncthreads();

  const int stride = gridDim.x * TPB;
  for (int i = blockIdx.x * TPB + tid; i < n4; i += stride) {
    if (i + stride < n4) {                 // gfx1250: global_prefetch_b8
      __builtin_prefetch(&pred4[i + stride], 0, 1);
      __builtin_prefetch(&tgt4[i + stride], 0, 1);
      __builtin_prefetch(&lw4[i + stride], 0, 1);
    }
    const float4 x = pred4[i];
    const float4 t = tgt4[i];
    const float4 w = lw4[i];
    ghmc_elem(x.x, t.x, w.x, base);
    ghmc_elem(x.y, t.y, w.y, base);
    ghmc_elem(x.z, t.z, w.z, base);
    ghmc_elem(x.w, t.w, w.w, base);
  }
  // scalar tail (n % 4), handled by block 0 only
  if (blockIdx.x == 0) {
    for (int i = n_tail_begin + tid; i < n; i += TPB)
      ghmc_elem(pred_s[i], tgt_s[i], lw_s[i], base);
  }
  __syncthreads();

  // Block reduction: 20 live slots, fixed summation order => deterministic.
  if (tid < 32) {
    float acc = 0.0f;
    if (tid < NSLOT)
      for (int wv = 0; wv < WAVES_PER_BLOCK; ++wv)
        for (int l = 0; l < 32; ++l)
          acc += hist[wv][tid][l];
    partials[(size_t)blockIdx.x * 32 + tid] = acc;   // slots 20..31 -> 0
  }
}

// ---------------------------------------------------------------------------
// Pass 2 (one block, 8 waves): column-sum the [nb x 32] partials matrix as
// onesT x P using chained V_WMMA_F32_16X16X4_F32 (f32 accumulate, exact for
// the integer-valued counts). With an all-ones A, any K-slot permutation of
// B's hardware layout yields identical column sums, so this is layout-robust.
// ---------------------------------------------------------------------------
__global__ void __launch_bounds__(TPB) ghmc_finalize_kernel(
    const float* __restrict__ partials, float* __restrict__ out, int nb) {
  const int tid = threadIdx.x;
  const int lane = tid & 31;
  const int wave = tid >> 5;
  const int rowsPerWave = nb / WAVES_PER_BLOCK;    // nb is a multiple of 32
  const int rb0 = wave * rowsPerWave;
  const int half = lane >> 4;
  const int col = lane & 15;

  v2f a; a[0] = 1.0f; a[1] = 1.0f;                 // A = ones (16x4 f32)
  v8f acc0 = {};                                   // cols 0..15 (slots 0..15)
  v8f acc1 = {};                                   // cols 16..31 (slots 16..19 + pad)

  for (int r = rb0; r < rb0 + rowsPerWave; r += 4) {
    const float* rowA = partials + (size_t)(r + half) * 32;
    const float* rowB = partials + (size_t)(r + 2 + half) * 32;
    v2f b0; b0[0] = rowA[col];      b0[1] = rowB[col];
    v2f b1; b1[0] = rowA[16 + col]; b1[1] = rowB[16 + col];
    acc0 = __builtin_amdgcn_wmma_f32_16x16x4_f32(false, a, false, b0,
                                                 (short)0, acc0, false, false);
    acc1 = __builtin_amdgcn_wmma_f32_16x16x4_f32(false, a, false, b1,
                                                 (short)0, acc1, false, false);
  }

  // D row 0 lives in accN[0] on lanes 0..15 (N = lane); all D rows are equal.
  __shared__ float partC[WAVES_PER_BLOCK][32];
  if (lane < 16) {
    partC[wave][lane] = acc0[0];
    partC[wave][16 + lane] = acc1[0];
  }
  __syncthreads();

  if (tid == 0) {
    float slot[NSLOT];
    for (int j = 0; j < NSLOT; ++j) {
      float s = 0.0f;
      for (int wv = 0; wv < WAVES_PER_BLOCK; ++wv) s += partC[wv][j];
      slot[j] = s;
    }
    float n_bins = 0.0f, lsum = 0.0f;
    for (int b = 0; b < GHMC_BINS; ++b) {
      const float cnt = slot[2 * b];
      if (cnt > 0.0f) {
        n_bins += 1.0f;
        lsum += slot[2 * b + 1] / cnt;
      }
    }
    out[0] = lsum / fmaxf(n_bins, 1.0f);           // LOSS_WEIGHT == 1.0
  }
}

extern "C" void kernel_launch(void* const* d_in, const int* in_sizes, int n_in,
                              void* d_out, int out_size, void* d_ws, size_t ws_size,
                              hipStream_t stream) {
  const float* pred = (const float*)d_in[0];
  const float* tgt  = (const float*)d_in[1];
  const float* lw   = (const float*)d_in[2];
  float* out = (float*)d_out;

  const int n  = in_sizes[0];          // 16,384,000
  const int n4 = n / 4;

  // nb blocks: multiple of 32 (finalize needs nb/8 divisible by 4), fits ws.
  int nb = 2048;
  const size_t per_block = 32 * sizeof(float);
  if (ws_size < (size_t)nb * per_block) {
    int cap = (int)(ws_size / per_block) & ~31;
    nb = cap < 32 ? 32 : cap;
  }
  float* partials = (float*)d_ws;

  ghmc_hist_kernel<<<nb, TPB, 0, stream>>>(
      (const float4*)pred, (const float4*)tgt, (const float4*)lw, partials,
      pred, tgt, lw, n4, n4 * 4, n);
  ghmc_finalize_kernel<<<1, TPB, 0, stream>>>(partials, out, nb);
}